// ReLU_Conv_20744692039895
// MI455X (gfx1250) — compile-verified
//
#include <hip/hip_runtime.h>

// Problem constants (C,H,W,hi,wi) = (32,32,32,32,32)
#define P_TOTAL   32768          // C*H*W pixels
#define TILE      1024           // hi*wi elements per pixel tile (contiguous)
#define NPIX      4              // pixels per wave (pipelined)
#define BLOCK     128            // 4 waves per block
#define WAVES_PB  (BLOCK / 32)

// Per-wave LDS: 2 buffers x (lx tile + ux tile) = 2*2*1024 floats = 16 KB
#define WAVE_LDS_FLOATS (2 * 2 * TILE)
#define BLOCK_LDS_FLOATS (WAVE_LDS_FLOATS * WAVES_PB)   // 16384 floats = 64 KB

typedef __attribute__((ext_vector_type(4))) float f4;

template <int N>
__device__ __forceinline__ void wait_async() {
    asm volatile("s_wait_asynccnt %0" :: "n"(N) : "memory");
}

// Issue one 4 KB tile (1024 f32) as 8 async b128 loads, global -> LDS.
// Lane L, chunk c covers elements [c*128 + 4L, c*128 + 4L + 4).
__device__ __forceinline__ void async_load_tile(const float* __restrict__ g,
                                                unsigned lds_byte, int lane) {
#pragma unroll
    for (int c = 0; c < 8; ++c) {
        const float* gp = g + c * 128 + lane * 4;
        unsigned     lp = lds_byte + (unsigned)(c * 512 + lane * 16);
        asm volatile("global_load_async_to_lds_b128 %0, %1, off"
                     :: "v"(lp), "v"(gp)
                     : "memory");
    }
}

__global__ __launch_bounds__(BLOCK) void relu_bounds_kernel(
    const float* __restrict__ lx_in, const float* __restrict__ ux_in,
    const float* __restrict__ lc_in, const float* __restrict__ uc_in,
    const float* __restrict__ x_min, const float* __restrict__ x_max,
    float* __restrict__ lx_out, float* __restrict__ ux_out,
    float* __restrict__ lc_out, float* __restrict__ uc_out)
{
    __shared__ float smem[BLOCK_LDS_FLOATS];

    const int lane = threadIdx.x & 31;
    const int wib  = threadIdx.x >> 5;                       // wave in block
    const int wave = blockIdx.x * WAVES_PB + wib;
    const int p0   = wave * NPIX;
    if (p0 >= P_TOTAL) return;                               // wave-uniform

    float*   wbuf      = &smem[wib * WAVE_LDS_FLOATS];
    unsigned wbuf_byte = (unsigned)(size_t)wbuf;             // LDS byte offset (low 32b of flat addr)

    // x_min / x_max slices for this lane -> VGPRs (reused for all pixels)
    const f4* xmin4 = (const f4*)x_min;
    const f4* xmax4 = (const f4*)x_max;
    f4 mn[8], mx[8];
#pragma unroll
    for (int c = 0; c < 8; ++c) {
        mn[c] = xmin4[c * 32 + lane];
        mx[c] = xmax4[c * 32 + lane];
    }

    // ---- pipeline prologue: fill both buffers (batches 0 and 1) ----
    async_load_tile(lx_in + (size_t)(p0 + 0) * TILE, wbuf_byte + 0 * 8192,        lane);
    async_load_tile(ux_in + (size_t)(p0 + 0) * TILE, wbuf_byte + 0 * 8192 + 4096, lane);
    async_load_tile(lx_in + (size_t)(p0 + 1) * TILE, wbuf_byte + 1 * 8192,        lane);
    async_load_tile(ux_in + (size_t)(p0 + 1) * TILE, wbuf_byte + 1 * 8192 + 4096, lane);

#pragma unroll
    for (int i = 0; i < NPIX; ++i) {
        const int b = i & 1;
        const int p = p0 + i;

        // Batch i complete when outstanding <= 16*(batches issued after i).
        if (i < NPIX - 1) wait_async<16>();
        else              wait_async<0>();

        const f4* lxb = (const f4*)&wbuf[b * 2048];
        const f4* uxb = (const f4*)&wbuf[b * 2048 + 1024];

        // ---- pass 1: reduction from LDS ----
        float lacc = 0.f, uacc = 0.f;
#pragma unroll
        for (int c = 0; c < 8; ++c) {
            f4 a = lxb[c * 32 + lane];
            f4 q = uxb[c * 32 + lane];
            // mask_lower*lx = max(lx,0)*x_min + min(lx,0)*x_max
            lacc += fmaxf(a.x, 0.f) * mn[c].x + fminf(a.x, 0.f) * mx[c].x;
            lacc += fmaxf(a.y, 0.f) * mn[c].y + fminf(a.y, 0.f) * mx[c].y;
            lacc += fmaxf(a.z, 0.f) * mn[c].z + fminf(a.z, 0.f) * mx[c].z;
            lacc += fmaxf(a.w, 0.f) * mn[c].w + fminf(a.w, 0.f) * mx[c].w;
            // mask_upper*ux = max(ux,0)*x_max + min(ux,0)*x_min
            uacc += fmaxf(q.x, 0.f) * mx[c].x + fminf(q.x, 0.f) * mn[c].x;
            uacc += fmaxf(q.y, 0.f) * mx[c].y + fminf(q.y, 0.f) * mn[c].y;
            uacc += fmaxf(q.z, 0.f) * mx[c].z + fminf(q.z, 0.f) * mn[c].z;
            uacc += fmaxf(q.w, 0.f) * mx[c].w + fminf(q.w, 0.f) * mn[c].w;
        }
        // wave32 butterfly reduce
#pragma unroll
        for (int off = 16; off > 0; off >>= 1) {
            lacc += __shfl_xor(lacc, off, 32);
            uacc += __shfl_xor(uacc, off, 32);
        }

        const float lc = lc_in[p];
        const float uc = uc_in[p];
        const float l = lacc + lc;
        const float u = uacc + uc;
        const bool alive = (l >= 0.f);
        const bool cross = (l < 0.f) && (u > 0.f);
        float slope = cross ? (u / (u - l)) : 1.f;
        slope = fminf(fmaxf(slope, 0.f), 1.f);
        const float lscale = alive ? 1.f : 0.f;
        const float uscale = alive ? 1.f : (cross ? slope : 0.f);

        // ---- pass 2: rescale tiles from LDS, stream to HBM (non-temporal) ----
        f4* lo = (f4*)(lx_out + (size_t)p * TILE);
        f4* uo = (f4*)(ux_out + (size_t)p * TILE);
#pragma unroll
        for (int c = 0; c < 8; ++c) {
            f4 a = lxb[c * 32 + lane];
            f4 q = uxb[c * 32 + lane];
            f4 ao = a * lscale;
            f4 qo = q * uscale;
            __builtin_nontemporal_store(ao, &lo[c * 32 + lane]);
            __builtin_nontemporal_store(qo, &uo[c * 32 + lane]);
        }
        if (lane == 0) {
            lc_out[p] = alive ? lc : 0.f;
            uc_out[p] = alive ? uc : (cross ? (slope * uc - slope * l) : 0.f);
        }

        // refill the buffer we just drained with batch i+2
        if (i + 2 < NPIX) {
            const int pn = p0 + i + 2;
            async_load_tile(lx_in + (size_t)pn * TILE, wbuf_byte + b * 8192,        lane);
            async_load_tile(ux_in + (size_t)pn * TILE, wbuf_byte + b * 8192 + 4096, lane);
        }
    }
}

extern "C" void kernel_launch(void* const* d_in, const int* in_sizes, int n_in,
                              void* d_out, int out_size, void* d_ws, size_t ws_size,
                              hipStream_t stream) {
    (void)in_sizes; (void)n_in; (void)d_ws; (void)ws_size; (void)out_size;
    const float* lx_in = (const float*)d_in[0];
    const float* ux_in = (const float*)d_in[1];
    const float* lc_in = (const float*)d_in[2];
    const float* uc_in = (const float*)d_in[3];
    const float* x_min = (const float*)d_in[4];
    const float* x_max = (const float*)d_in[5];

    float* out    = (float*)d_out;
    float* lx_out = out;                         // 32*32*32*32*32 = 33554432
    float* ux_out = out + 33554432;
    float* lc_out = out + 67108864;              // 2*33554432
    float* uc_out = out + 67141632;              // + 32768

    const int blocks = P_TOTAL / (NPIX * WAVES_PB);  // 32768 / 16 = 2048
    relu_bounds_kernel<<<blocks, BLOCK, 0, stream>>>(
        lx_in, ux_in, lc_in, uc_in, x_min, x_max,
        lx_out, ux_out, lc_out, uc_out);
}